// pool_layer_2190433321288
// MI455X (gfx1250) — compile-verified
//
#include <hip/hip_runtime.h>
#include <hip/hip_bf16.h>
#include <stdint.h>

// out[n,f,b] = (1/7) * sum_{k=0..6} x[ idx[n*7 + (7f+k)>>7], (7f+k)&127, b ]
// == per node: concat the 7 gathered rows as 896 float4s G[], then
//    out4[n,f] = mean(G[7f .. 7f+6]).
// Pure bandwidth problem (~672 MB total -> ~29us @ 23.3 TB/s). Strategy:
// async-copy (GLOBAL_LOAD_ASYNC_TO_LDS_B128) the 7 rows coalesced into LDS,
// absorb the stride-7 permutation in LDS, stream results out non-temporally.

typedef float v4f __attribute__((ext_vector_type(4)));
typedef int   v4i __attribute__((vector_size(16)));   // matches builtin param

#define LDS_AS __attribute__((address_space(3)))
#define GLB_AS __attribute__((address_space(1)))

__device__ __forceinline__ LDS_AS v4i* as_lds_v4(const void* p) {
    // generic LDS pointer keeps the LDS byte offset in its low 32 bits
    return (LDS_AS v4i*)(uint32_t)(uintptr_t)p;
}
__device__ __forceinline__ GLB_AS v4i* as_glb_v4(const void* p) {
    return (GLB_AS v4i*)(uintptr_t)p;
}

__global__ __launch_bounds__(128) void gather_mean7_kernel(
    const float* __restrict__ x,       // (N_IN, 128, 4) f32, row = 512 floats
    const int*   __restrict__ neigh,   // (n_out*7) indices
    float*       __restrict__ out,     // (n_out, 128, 4) f32
    int n_out)
{
    __shared__ v4f lds[7 * 128];       // 14 KB: 7 gathered rows x 128 float4

    const int n   = blockIdx.x;
    const int tid = threadIdx.x;       // 0..127
    if (n >= n_out) return;            // grid == n_out; uniform, never taken

    const int* no = neigh + (size_t)n * 7;

    // Phase 1: stage 7 rows into LDS, coalesced 512B per wave per issue.
#pragma unroll
    for (int c = 0; c < 7; ++c) {
        const long  row = (long)no[c];                 // block-uniform
        const float* src = x + row * 512 + tid * 4;    // float4 'tid' of row
        __builtin_amdgcn_global_load_async_to_lds_b128(
            as_glb_v4(src), as_lds_v4(&lds[c * 128 + tid]),
            /*imm offset*/0, /*cpol*/0);
    }

#if __has_builtin(__builtin_amdgcn_s_wait_asynccnt)
    __builtin_amdgcn_s_wait_asynccnt(0);
#else
    asm volatile("s_wait_asynccnt 0" ::: "memory");
#endif
    __syncthreads();   // make all waves' async copies visible to each other

    // Phase 2: out4[f=tid] = mean of 7 consecutive float4s.
    const v4f* g = (const v4f*)&lds[tid * 7];
    v4f s = g[0];
    s += g[1]; s += g[2]; s += g[3]; s += g[4]; s += g[5]; s += g[6];
    s *= (1.0f / 7.0f);

    v4f* o = (v4f*)out;
    __builtin_nontemporal_store(s, &o[(size_t)n * 128 + tid]);
}

extern "C" void kernel_launch(void* const* d_in, const int* in_sizes, int n_in,
                              void* d_out, int out_size, void* d_ws, size_t ws_size,
                              hipStream_t stream) {
    const float* x     = (const float*)d_in[0];
    const int*   neigh = (const int*)d_in[1];
    float*       out   = (float*)d_out;

    const int n_out = in_sizes[1] / 7;   // 40962 for the reference shapes

    dim3 grid(n_out), block(128);
    hipLaunchKernelGGL(gather_mean7_kernel, grid, block, 0, stream,
                       x, neigh, out, n_out);
}